// ClockAwareGNN_38354057953670
// MI455X (gfx1250) — compile-verified
//
#include <hip/hip_runtime.h>

#define NGRAPHS 1024
#define DFEAT   128
#define HID     16
#define CHUNK   2048

typedef __attribute__((ext_vector_type(2))) float v2f;
typedef __attribute__((ext_vector_type(4))) float v4f;
typedef __attribute__((ext_vector_type(8))) float v8f;

// ---------------- zero workspace accumulators ----------------
__global__ void k_zero(float* __restrict__ ws, int n) {
    int i = blockIdx.x * blockDim.x + threadIdx.x;
    if (i < n) ws[i] = 0.0f;
}

// ---------------- precompute folded weights ----------------
// Mmat[k][n] = sum_j W2[k][j] * W3[1+j][n]   (16 x 32)
// c3[n]     = b3[n] + sum_j b2[j] * W3[1+j][n]
__global__ void k_prep(const float* __restrict__ W2, const float* __restrict__ b2,
                       const float* __restrict__ W3, const float* __restrict__ b3,
                       float* __restrict__ Mmat, float* __restrict__ c3) {
    int t = threadIdx.x;
    if (t < 512) {
        int k = t >> 5, n = t & 31;
        float s = 0.0f;
        #pragma unroll
        for (int j = 0; j < HID; ++j) s += W2[k * HID + j] * W3[(1 + j) * 32 + n];
        Mmat[t] = s;
    } else if (t < 544) {
        int n = t - 512;
        float s = b3[n];
        #pragma unroll
        for (int j = 0; j < HID; ++j) s += b2[j] * W3[(1 + j) * 32 + n];
        c3[n] = s;
    }
}

// ---------------- main streaming pass ----------------
// 256 threads: 8 node-rows (r = tid>>5) x 32 feature-groups of 4 (f4 = tid&31).
// Register accumulation per (segment, feature); flush on segment change.
// x / clock_period are streamed once (1 GB >> 192 MB L2) -> non-temporal loads
// so the stream does not evict the hot atomic-accumulator lines.
__device__ __forceinline__ void flush_seg(int cur, int f4,
                                          v4f& ax, v4f& aa, float& ac,
                                          float* __restrict__ sum_x,
                                          float* __restrict__ cnt,
                                          float* __restrict__ sum_a) {
    if (cur >= 0) {
        float* sx = sum_x + (size_t)cur * DFEAT + f4 * 4;
        atomicAdd(sx + 0, ax.x); atomicAdd(sx + 1, ax.y);
        atomicAdd(sx + 2, ax.z); atomicAdd(sx + 3, ax.w);
        if (f4 < 4) {
            float* sa = sum_a + cur * HID + f4 * 4;
            atomicAdd(sa + 0, aa.x); atomicAdd(sa + 1, aa.y);
            atomicAdd(sa + 2, aa.z); atomicAdd(sa + 3, aa.w);
        }
        if (f4 == 0) atomicAdd(cnt + cur, ac);
    }
    ax = (v4f){0.f, 0.f, 0.f, 0.f};
    aa = (v4f){0.f, 0.f, 0.f, 0.f};
    ac = 0.f;
}

__global__ void __launch_bounds__(256)
k_main(const v4f* __restrict__ x4, const float* __restrict__ cp,
       const int* __restrict__ batch,
       const float* __restrict__ W1, const float* __restrict__ b1,
       float* __restrict__ sum_x, float* __restrict__ cnt,
       float* __restrict__ sum_a, int N) {
    const int t  = threadIdx.x;
    const int f4 = t & 31;   // features f4*4 .. f4*4+3
    const int r  = t >> 5;   // node row 0..7
    const int base = blockIdx.x * CHUNK;
    int end = base + CHUNK; if (end > N) end = N;

    v4f w1v = (v4f){0.f, 0.f, 0.f, 0.f};
    v4f b1v = (v4f){0.f, 0.f, 0.f, 0.f};
    if (f4 < 4) {
        w1v = ((const v4f*)W1)[f4];
        b1v = ((const v4f*)b1)[f4];
    }

    v4f  ax = (v4f){0.f, 0.f, 0.f, 0.f};
    v4f  aa = (v4f){0.f, 0.f, 0.f, 0.f};
    float ac = 0.f;
    int cur = -1;

    for (int n = base + r; n < end; n += 8) {
        int seg = batch[n];                       // wave-uniform (same n per wave)
        if (seg != cur) {
            flush_seg(cur, f4, ax, aa, ac, sum_x, cnt, sum_a);
            cur = seg;
        }
        // streamed exactly once -> non-temporal (TH=NT) 512B/wave32 load
        v4f xv = __builtin_nontemporal_load(x4 + (size_t)n * 32 + f4);
        ax.x += xv.x; ax.y += xv.y; ax.z += xv.z; ax.w += xv.w;
        if (f4 < 4) {
            float c = __builtin_nontemporal_load(cp + n);
            aa.x += fmaxf(c * w1v.x + b1v.x, 0.f);
            aa.y += fmaxf(c * w1v.y + b1v.y, 0.f);
            aa.z += fmaxf(c * w1v.z + b1v.z, 0.f);
            aa.w += fmaxf(c * w1v.w + b1v.w, 0.f);
        }
        if (f4 == 0) ac += 1.f;
    }
    flush_seg(cur, f4, ax, aa, ac, sum_x, cnt, sum_a);
}

// ---------------- graph-level epilogue: WMMA f32 16x16x4 ----------------
// One wave handles 16 graphs: D[16x32] = mean_a[16x16] @ Mmat[16x32]
// via 4 K-steps x 2 N-tiles of V_WMMA_F32_16X16X4_F32 (exact fp32).
__global__ void __launch_bounds__(32)
k_final(const float* __restrict__ sum_x, const float* __restrict__ cnt,
        const float* __restrict__ sum_a, const float* __restrict__ Mmat,
        const float* __restrict__ c3,
        const float* __restrict__ W_base, const float* __restrict__ b_base,
        const float* __restrict__ W3, const float* __restrict__ W4,
        const float* __restrict__ b4, float* __restrict__ out) {
    __shared__ float tile[16][32];
    const int lane = threadIdx.x;
    const int half = lane >> 4;      // 0: K/M-low group, 1: high
    const int lm   = lane & 15;
    const int g0   = blockIdx.x * 16;

    // A fragment: lane holds mean_a row M=lm, columns K = s*4 + half*2 + {0,1}
    const float invc = 1.0f / fmaxf(cnt[g0 + lm], 1.0f);
    const float* sa = sum_a + (size_t)(g0 + lm) * HID;

    v8f acc0 = {0.f,0.f,0.f,0.f,0.f,0.f,0.f,0.f};
    v8f acc1 = {0.f,0.f,0.f,0.f,0.f,0.f,0.f,0.f};
    #pragma unroll
    for (int s = 0; s < 4; ++s) {
        const int k = s * 4 + half * 2;
        v2f a;  a.x  = sa[k] * invc;            a.y  = sa[k + 1] * invc;
        v2f b0; b0.x = Mmat[k * 32 + lm];       b0.y = Mmat[(k + 1) * 32 + lm];
        v2f bb; bb.x = Mmat[k * 32 + 16 + lm];  bb.y = Mmat[(k + 1) * 32 + 16 + lm];
        acc0 = __builtin_amdgcn_wmma_f32_16x16x4_f32(
                   false, a, false, b0, (short)0, acc0, false, false);
        acc1 = __builtin_amdgcn_wmma_f32_16x16x4_f32(
                   false, a, false, bb, (short)0, acc1, false, false);
    }

    // C/D layout: VGPR v, lane -> (M = v + 8*half, N = lm [+16 for tile 1])
    #pragma unroll
    for (int v = 0; v < 8; ++v) {
        tile[v + 8 * half][lm]      = acc0[v];
        tile[v + 8 * half][16 + lm] = acc1[v];
    }
    __syncthreads();

    if (lane < 16) {
        const int g = g0 + lane;
        const float ic = 1.0f / fmaxf(cnt[g], 1.0f);
        const float* sx = sum_x + (size_t)g * DFEAT;
        float gp = 0.0f;
        #pragma unroll 4
        for (int j = 0; j < DFEAT; ++j) gp += sx[j] * W_base[j];
        gp = gp * ic + b_base[0];

        float o = b4[0];
        #pragma unroll
        for (int n = 0; n < 32; ++n) {
            float val = tile[lane][n] + gp * W3[n] + c3[n];  // W3 row 0 = graph_pred weights
            o += fmaxf(val, 0.0f) * W4[n];
        }
        out[g] = o;
    }
}

// ---------------- host launch ----------------
extern "C" void kernel_launch(void* const* d_in, const int* in_sizes, int n_in,
                              void* d_out, int out_size, void* d_ws, size_t ws_size,
                              hipStream_t stream) {
    const float* x      = (const float*)d_in[0];
    const float* cp     = (const float*)d_in[1];
    const int*   batch  = (const int*)d_in[2];
    const float* W_base = (const float*)d_in[3];
    const float* b_base = (const float*)d_in[4];
    const float* W1     = (const float*)d_in[5];
    const float* b1     = (const float*)d_in[6];
    const float* W2     = (const float*)d_in[7];
    const float* b2     = (const float*)d_in[8];
    const float* W3     = (const float*)d_in[9];
    const float* b3     = (const float*)d_in[10];
    const float* W4     = (const float*)d_in[11];
    const float* b4     = (const float*)d_in[12];
    float* out = (float*)d_out;
    float* ws  = (float*)d_ws;

    const int N = in_sizes[0] / DFEAT;   // 2,000,000

    float* sum_x = ws;                                   // [1024*128]
    float* cnt   = sum_x + (size_t)NGRAPHS * DFEAT;      // [1024]
    float* sum_a = cnt + NGRAPHS;                        // [1024*16]
    float* Mmat  = sum_a + (size_t)NGRAPHS * HID;        // [16*32]
    float* c3    = Mmat + HID * 32;                      // [32]

    const int zn = NGRAPHS * DFEAT + NGRAPHS + NGRAPHS * HID;
    k_zero<<<(zn + 255) / 256, 256, 0, stream>>>(ws, zn);
    k_prep<<<1, 544, 0, stream>>>(W2, b2, W3, b3, Mmat, c3);

    const int nblk = (N + CHUNK - 1) / CHUNK;
    k_main<<<nblk, 256, 0, stream>>>((const v4f*)x, cp, batch, W1, b1,
                                     sum_x, cnt, sum_a, N);
    k_final<<<NGRAPHS / 16, 32, 0, stream>>>(sum_x, cnt, sum_a, Mmat, c3,
                                             W_base, b_base, W3, W4, b4, out);
}